// MoE_65060164600307
// MI455X (gfx1250) — compile-verified
//
#include <hip/hip_runtime.h>

typedef __attribute__((ext_vector_type(2))) float v2f;
typedef __attribute__((ext_vector_type(8))) float v8f;

#define DIM   1024
#define INTER 512
#define NEXP  8
#define KC    64           // K chunk staged in LDS
#define LSTR  (KC + 4)     // padded LDS row stride (dwords): conflict-free b64 reads

__device__ __forceinline__ float silu_f(float g) {
    return g / (1.0f + __expf(-g));
}

__device__ __forceinline__ v8f wmma_f32(v2f a, v2f b, v8f c) {
    return __builtin_amdgcn_wmma_f32_16x16x4_f32(
        false, a, false, b, (short)0, c, false, false);
}

// ---------------- Gate: softmax(x @ gate_w^T), top-2, dense combine weights ----
__global__ __launch_bounds__(256) void moe_gate_kernel(
    const float* __restrict__ X, const float* __restrict__ GW,
    float* __restrict__ CW, int N)
{
    int lane = threadIdx.x & 31;
    int tok  = (blockIdx.x * blockDim.x + threadIdx.x) >> 5;
    if (tok >= N) return;
    const float* xr = X + (size_t)tok * DIM;
    float acc[NEXP];
#pragma unroll
    for (int e = 0; e < NEXP; ++e) acc[e] = 0.0f;
    for (int d = lane; d < DIM; d += 32) {
        float xv = xr[d];
#pragma unroll
        for (int e = 0; e < NEXP; ++e) acc[e] += xv * GW[e * DIM + d];
    }
#pragma unroll
    for (int e = 0; e < NEXP; ++e) {
#pragma unroll
        for (int off = 16; off > 0; off >>= 1)
            acc[e] += __shfl_xor(acc[e], off, 32);
    }
    if (lane == 0) {
        float mx = acc[0];
#pragma unroll
        for (int e = 1; e < NEXP; ++e) mx = fmaxf(mx, acc[e]);
        float s[NEXP]; float sum = 0.0f;
#pragma unroll
        for (int e = 0; e < NEXP; ++e) { s[e] = __expf(acc[e] - mx); sum += s[e]; }
        float inv = 1.0f / sum;
#pragma unroll
        for (int e = 0; e < NEXP; ++e) s[e] *= inv;
        int i1 = 0;
#pragma unroll
        for (int e = 1; e < NEXP; ++e) if (s[e] > s[i1]) i1 = e;
        int i2 = (i1 == 0) ? 1 : 0;
#pragma unroll
        for (int e = 0; e < NEXP; ++e) if (e != i1 && s[e] > s[i2]) i2 = e;
#pragma unroll
        for (int e = 0; e < NEXP; ++e)
            CW[(size_t)tok * NEXP + e] = (e == i1) ? s[i1] : ((e == i2) ? s[i2] : 0.0f);
    }
}

// ---------------- GEMM1: H = silu(X Wg^T) * (X Wu^T), one expert ---------------
// Block: 8 waves, 32 token rows x 512 (all INTER) cols. A tile staged in LDS.
// Wave: 32(M) x 64(N) strip = 2x4 WMMA tiles for G and for U.
__global__ __launch_bounds__(256) void moe_gemm1_kernel(
    const float* __restrict__ X,    // [N, DIM]
    const float* __restrict__ Wg,   // [INTER, DIM]
    const float* __restrict__ Wu,   // [INTER, DIM]
    float* __restrict__ H,          // [N, INTER]
    int N)
{
    __shared__ float lx[32 * LSTR];

    int tid  = threadIdx.x;
    int lane = tid & 31;
    int wv   = tid >> 5;            // 0..7
    int m0   = blockIdx.x * 32;
    if (m0 >= N) return;            // uniform
    int n0   = wv * 64;

    int kHalf = (lane >> 4) << 1;   // 0 or 2
    int mr    = lane & 15;
    int nc    = lane & 15;

    const float* bgRow[4];
    const float* buRow[4];
#pragma unroll
    for (int t = 0; t < 4; ++t) {
        int n = n0 + t * 16 + nc;
        bgRow[t] = Wg + (size_t)n * DIM + kHalf;
        buRow[t] = Wu + (size_t)n * DIM + kHalf;
    }

    v8f zero = {};
    v8f accG[2][4], accU[2][4];
#pragma unroll
    for (int mt = 0; mt < 2; ++mt)
#pragma unroll
        for (int t = 0; t < 4; ++t) { accG[mt][t] = zero; accU[mt][t] = zero; }

    for (int kc = 0; kc < DIM; kc += KC) {
        __syncthreads();
        // Stage X[m0:m0+32, kc:kc+KC] -> LDS (512 float4, 2 per thread, coalesced)
#pragma unroll
        for (int i = tid; i < 512; i += 256) {
            int r  = i >> 4;        // 0..31
            int c4 = i & 15;        // 0..15
            *(float4*)&lx[r * LSTR + c4 * 4] =
                *(const float4*)&X[(size_t)(m0 + r) * DIM + kc + c4 * 4];
        }
        __syncthreads();

        for (int k = 0; k < KC; k += 4) {
            int kg = kc + k;
            v2f a0 = *(const v2f*)&lx[mr * LSTR + k + kHalf];
            v2f a1 = *(const v2f*)&lx[(16 + mr) * LSTR + k + kHalf];
#pragma unroll
            for (int t = 0; t < 4; ++t) {
                v2f bg = *(const v2f*)(bgRow[t] + kg);
                v2f bu = *(const v2f*)(buRow[t] + kg);
                accG[0][t] = wmma_f32(a0, bg, accG[0][t]);
                accU[0][t] = wmma_f32(a0, bu, accU[0][t]);
                accG[1][t] = wmma_f32(a1, bg, accG[1][t]);
                accU[1][t] = wmma_f32(a1, bu, accU[1][t]);
            }
        }
    }

    int jBase = (lane >> 4) << 3;   // 0 or 8
#pragma unroll
    for (int mt = 0; mt < 2; ++mt) {
#pragma unroll
        for (int t = 0; t < 4; ++t) {
#pragma unroll
            for (int j = 0; j < 8; ++j) {
                int row = m0 + mt * 16 + jBase + j;
                float g = accG[mt][t][j];
                float u = accU[mt][t][j];
                H[(size_t)row * INTER + n0 + t * 16 + nc] = silu_f(g) * u;
            }
        }
    }
}

// ---------------- GEMM2: Out (+)= diag(w) * (H Wd^T), one expert ---------------
// Block: 8 waves, 32 rows x 512 cols (grid.y spans DIM/512). A tile via LDS.
__global__ __launch_bounds__(256) void moe_gemm2_kernel(
    const float* __restrict__ H,    // [N, INTER]
    const float* __restrict__ Wd,   // [DIM, INTER]
    const float* __restrict__ CW,   // [N, NEXP]
    int expert,                     // 0..7 routed, -1 shared (weight 1.0)
    int accumulate,                 // 0 = overwrite, 1 = add
    float* __restrict__ Out,        // [N, DIM]
    int N)
{
    __shared__ float lh[32 * LSTR];

    int tid  = threadIdx.x;
    int lane = tid & 31;
    int wv   = tid >> 5;
    int m0   = blockIdx.x * 32;
    if (m0 >= N) return;
    int n0   = blockIdx.y * 512 + wv * 64;

    int kHalf = (lane >> 4) << 1;
    int mr    = lane & 15;
    int nc    = lane & 15;

    const float* bRow[4];
#pragma unroll
    for (int t = 0; t < 4; ++t)
        bRow[t] = Wd + (size_t)(n0 + t * 16 + nc) * INTER + kHalf;

    v8f zero = {};
    v8f acc[2][4];
#pragma unroll
    for (int mt = 0; mt < 2; ++mt)
#pragma unroll
        for (int t = 0; t < 4; ++t) acc[mt][t] = zero;

    for (int kc = 0; kc < INTER; kc += KC) {
        __syncthreads();
#pragma unroll
        for (int i = tid; i < 512; i += 256) {
            int r  = i >> 4;
            int c4 = i & 15;
            *(float4*)&lh[r * LSTR + c4 * 4] =
                *(const float4*)&H[(size_t)(m0 + r) * INTER + kc + c4 * 4];
        }
        __syncthreads();

        for (int k = 0; k < KC; k += 4) {
            int kg = kc + k;
            v2f a0 = *(const v2f*)&lh[mr * LSTR + k + kHalf];
            v2f a1 = *(const v2f*)&lh[(16 + mr) * LSTR + k + kHalf];
#pragma unroll
            for (int t = 0; t < 4; ++t) {
                v2f b = *(const v2f*)(bRow[t] + kg);
                acc[0][t] = wmma_f32(a0, b, acc[0][t]);
                acc[1][t] = wmma_f32(a1, b, acc[1][t]);
            }
        }
    }

    int jBase = (lane >> 4) << 3;
#pragma unroll
    for (int mt = 0; mt < 2; ++mt) {
        float w[8];
#pragma unroll
        for (int j = 0; j < 8; ++j) {
            int row = m0 + mt * 16 + jBase + j;
            w[j] = (expert >= 0) ? CW[(size_t)row * NEXP + expert] : 1.0f;
        }
#pragma unroll
        for (int t = 0; t < 4; ++t) {
#pragma unroll
            for (int j = 0; j < 8; ++j) {
                int row = m0 + mt * 16 + jBase + j;
                size_t idx = (size_t)row * DIM + n0 + t * 16 + nc;
                float v = w[j] * acc[mt][t][j];
                if (accumulate) Out[idx] += v;
                else            Out[idx] = v;
            }
        }
    }
}

extern "C" void kernel_launch(void* const* d_in, const int* in_sizes, int n_in,
                              void* d_out, int out_size, void* d_ws, size_t ws_size,
                              hipStream_t stream) {
    const float* x      = (const float*)d_in[0];
    const float* gate_w = (const float*)d_in[1];
    const float* wg     = (const float*)d_in[2];
    const float* wu     = (const float*)d_in[3];
    const float* wd     = (const float*)d_in[4];
    const float* sg     = (const float*)d_in[5];
    const float* su     = (const float*)d_in[6];
    const float* sd     = (const float*)d_in[7];
    float* out = (float*)d_out;

    const int N = in_sizes[0] / DIM;                // 4096 tokens

    float* cw   = (float*)d_ws;                     // [N, NEXP]
    float* Hbuf = cw + (size_t)N * NEXP;            // [N, INTER]

    moe_gate_kernel<<<(N + 7) / 8, 256, 0, stream>>>(x, gate_w, cw, N);

    for (int e = 0; e < NEXP + 1; ++e) {
        const float* wgp = (e < NEXP) ? wg + (size_t)e * INTER * DIM : sg;
        const float* wup = (e < NEXP) ? wu + (size_t)e * INTER * DIM : su;
        const float* wdp = (e < NEXP) ? wd + (size_t)e * DIM * INTER : sd;

        moe_gemm1_kernel<<<N / 32, 256, 0, stream>>>(x, wgp, wup, Hbuf, N);

        dim3 g2(N / 32, DIM / 512);
        moe_gemm2_kernel<<<g2, 256, 0, stream>>>(
            Hbuf, wdp, cw, (e < NEXP) ? e : -1, (e > 0) ? 1 : 0, out, N);
    }
}